// PreNorm_317827580172
// MI455X (gfx1250) — compile-verified
//
#include <hip/hip_runtime.h>

typedef __attribute__((ext_vector_type(2))) float v2f;
typedef __attribute__((ext_vector_type(8))) float v8f;

#define N_DIM 4096
#define M_DIM 16
#define D_DIM 128
#define H_DIM 32
#define P_DIM 8192
#define CHUNKS 8
#define CHUNK_S 1024   // P_DIM / CHUNKS

__device__ __forceinline__ v8f wmma_f32(v2f a, v2f b, v8f c) {
    // D = A(16x4 f32) * B(4x16 f32) + C(16x16 f32)
    return __builtin_amdgcn_wmma_f32_16x16x4_f32(
        /*neg_a=*/false, a, /*neg_b=*/false, b,
        /*c_mod=*/(short)0, c, /*reuse_a=*/false, /*reuse_b=*/false);
}

__device__ __forceinline__ v8f v8f_zero() {
    v8f z = {0.f, 0.f, 0.f, 0.f, 0.f, 0.f, 0.f, 0.f};
    return z;
}

// ---------------------------------------------------------------------------
// Kernel 1: RMSNorm (no eps, no learned scale) : Xn = X * rsqrt(mean(X^2))
// ---------------------------------------------------------------------------
__global__ void rmsnorm_kernel(const float* __restrict__ X, float* __restrict__ Xn) {
    __shared__ float red[256];
    const int row = blockIdx.x;
    const int tid = threadIdx.x;
    const float* xr = X + (size_t)row * N_DIM;
    float s = 0.f;
    for (int i = tid; i < N_DIM; i += 256) { float v = xr[i]; s += v * v; }
    red[tid] = s;
    __syncthreads();
    for (int off = 128; off > 0; off >>= 1) {
        if (tid < off) red[tid] += red[tid + off];
        __syncthreads();
    }
    const float rs = rsqrtf(red[0] * (1.0f / (float)N_DIM));
    float* onr = Xn + (size_t)row * N_DIM;
    for (int i = tid; i < N_DIM; i += 256) onr[i] = xr[i] * rs;
}

// ---------------------------------------------------------------------------
// Kernel 2: QKV projection.  [16,4096] x [4096,4096] -> [16,4096], x3.
// One wave per 16-column output tile, fp32 WMMA 16x16x4 over K=4096.
// grid = (64, 3), block = 128 (4 waves, 64 cols per block)
// ---------------------------------------------------------------------------
__global__ __launch_bounds__(128)
void qkv_kernel(const float* __restrict__ Xn,
                const float* __restrict__ Wq,
                const float* __restrict__ Wk,
                const float* __restrict__ Wv,
                float* __restrict__ Q, float* __restrict__ K, float* __restrict__ V) {
    const float* W = (blockIdx.y == 0) ? Wq : ((blockIdx.y == 1) ? Wk : Wv);
    float*       O = (blockIdx.y == 0) ? Q  : ((blockIdx.y == 1) ? K  : V);

    const int lane   = threadIdx.x & 31;
    const int wave   = threadIdx.x >> 5;
    const int col    = blockIdx.x * 64 + wave * 16;
    const int lane15 = lane & 15;
    const int hi     = lane >> 4;
    const int krow   = hi << 1;          // 0 or 2

    const float* ap = Xn + (size_t)lane15 * N_DIM + krow;            // A: row lane15
    const float* bp = W + (size_t)krow * N_DIM + col + lane15;        // B: col (col+lane15)

    v8f acc = v8f_zero();
#pragma unroll 8
    for (int j = 0; j < N_DIM / 4; ++j) {
        v2f a = *(const v2f*)(ap + 4 * j);
        const float* b0 = bp + (size_t)(4 * j) * N_DIM;
        v2f b;
        b.x = b0[0];
        b.y = b0[N_DIM];
        acc = wmma_f32(a, b, acc);
    }
    const int rowHi = hi * 8;
#pragma unroll
    for (int r = 0; r < 8; ++r)
        O[(size_t)(r + rowHi) * N_DIM + col + lane15] = acc[r];
}

// ---------------------------------------------------------------------------
// Kernel 3: split-S flash attention partials.
// grid = (CHUNKS, H), block = 32 (one wave).  Each wave: one (head, S-chunk).
// Chunk 7 additionally consumes the 16 appended K/V rows from the workspace.
// ---------------------------------------------------------------------------
__global__ __launch_bounds__(32)
void attn_partial_kernel(const float* __restrict__ Q,
                         const float* __restrict__ Kn,   // new K, [16, 4096]
                         const float* __restrict__ Vn,   // new V, [16, 4096]
                         const float* __restrict__ cacheK,
                         const float* __restrict__ cacheV,
                         float* __restrict__ Op, float* __restrict__ Mp,
                         float* __restrict__ Lp) {
    __shared__ float lds[16 * 20];       // 16x16 P tile, padded stride 20
    const int c      = blockIdx.x;
    const int h      = blockIdx.y;
    const int lane   = threadIdx.x & 31;
    const int lane15 = lane & 15;
    const int hi     = lane >> 4;
    const int krow   = hi << 1;

    // q for this head, held in registers: A-operand pairs for 32 k-steps (D=128)
    v2f qreg[32];
    const float* qp = Q + (size_t)lane15 * N_DIM + h * D_DIM + krow;
#pragma unroll
    for (int j = 0; j < 32; ++j) qreg[j] = *(const v2f*)(qp + 4 * j);

    float rowmax[8], rowsum[8];
    v8f o[8];
#pragma unroll
    for (int r = 0; r < 8; ++r) { rowmax[r] = -__builtin_inff(); rowsum[r] = 0.f; }
#pragma unroll
    for (int t = 0; t < 8; ++t) o[t] = v8f_zero();

    const int ntiles = (c == CHUNKS - 1) ? (CHUNK_S / 16 + 1) : (CHUNK_S / 16);
    for (int t = 0; t < ntiles; ++t) {
        const float* Kt;
        const float* Vt;
        size_t str;
        if (t < CHUNK_S / 16) {
            const size_t sBase = (size_t)h * P_DIM + (size_t)c * CHUNK_S + (size_t)t * 16;
            Kt  = cacheK + sBase * D_DIM;
            Vt  = cacheV + sBase * D_DIM;
            str = D_DIM;
        } else {                         // the 16 freshly-appended rows
            Kt  = Kn + h * D_DIM;
            Vt  = Vn + h * D_DIM;
            str = N_DIM;
        }

        // ---- scores tile S = q (16x128) . K^T (128x16) -------------------
        v8f s = v8f_zero();
        const float* kp = Kt + (size_t)lane15 * str + krow;
#pragma unroll
        for (int j = 0; j < 32; ++j) {
            v2f b = *(const v2f*)(kp + 4 * j);
            s = wmma_f32(qreg[j], b, s);
        }

        // ---- online softmax update (per-row reductions over 16-lane halves)
        float scale[8], pv[8];
#pragma unroll
        for (int r = 0; r < 8; ++r) {
            float m = s[r];
            m = fmaxf(m, __shfl_xor(m, 1, 32));
            m = fmaxf(m, __shfl_xor(m, 2, 32));
            m = fmaxf(m, __shfl_xor(m, 4, 32));
            m = fmaxf(m, __shfl_xor(m, 8, 32));
            const float mn = fmaxf(rowmax[r], m);
            scale[r] = __expf(rowmax[r] - mn);
            rowmax[r] = mn;
            const float p = __expf(s[r] - mn);
            pv[r] = p;
            float ls = p;
            ls += __shfl_xor(ls, 1, 32);
            ls += __shfl_xor(ls, 2, 32);
            ls += __shfl_xor(ls, 4, 32);
            ls += __shfl_xor(ls, 8, 32);
            rowsum[r] = rowsum[r] * scale[r] + ls;
        }
#pragma unroll
        for (int tt = 0; tt < 8; ++tt)
#pragma unroll
            for (int r = 0; r < 8; ++r)
                o[tt][r] *= scale[r];

        // ---- P: C-layout -> A-layout via LDS ----------------------------
#pragma unroll
        for (int r = 0; r < 8; ++r)
            lds[(r + hi * 8) * 20 + lane15] = pv[r];
        asm volatile("s_wait_dscnt 0" ::: "memory");
        v2f pa[4];
#pragma unroll
        for (int j = 0; j < 4; ++j)
            pa[j] = *(const v2f*)(&lds[lane15 * 20 + 4 * j + krow]);

        // ---- O += P (16x16) . V (16x128) --------------------------------
#pragma unroll
        for (int tt = 0; tt < 8; ++tt) {
            const float* vp = Vt + tt * 16 + lane15;
#pragma unroll
            for (int j = 0; j < 4; ++j) {
                const float* v0 = vp + (size_t)(4 * j + krow) * str;
                v2f b;
                b.x = v0[0];
                b.y = v0[str];
                o[tt] = wmma_f32(pa[j], b, o[tt]);
            }
        }
    }

    // ---- write partials (unnormalized O, row max m, row sum l) ----------
    const size_t pb = ((size_t)h * CHUNKS + c) * 16;
#pragma unroll
    for (int tt = 0; tt < 8; ++tt)
#pragma unroll
        for (int r = 0; r < 8; ++r)
            Op[(pb + r + hi * 8) * D_DIM + tt * 16 + lane15] = o[tt][r];
#pragma unroll
    for (int r = 0; r < 8; ++r) {
        if (lane15 == r) {
            Mp[pb + r + hi * 8] = rowmax[r];
            Lp[pb + r + hi * 8] = rowsum[r];
        }
    }
}

// ---------------------------------------------------------------------------
// Kernel 4: merge the 8 chunk-partials per head, write [16, 4096] output.
// grid = H, block = 128 (one thread per d)
// ---------------------------------------------------------------------------
__global__ void combine_kernel(const float* __restrict__ Op,
                               const float* __restrict__ Mp,
                               const float* __restrict__ Lp,
                               float* __restrict__ out) {
    const int h = blockIdx.x;
    const int d = threadIdx.x;   // 0..127
    for (int row = 0; row < M_DIM; ++row) {
        float M = -__builtin_inff();
        for (int cc = 0; cc < CHUNKS; ++cc)
            M = fmaxf(M, Mp[((size_t)h * CHUNKS + cc) * 16 + row]);
        float L = 0.f, acc = 0.f;
        for (int cc = 0; cc < CHUNKS; ++cc) {
            const size_t idx = ((size_t)h * CHUNKS + cc) * 16 + row;
            const float w = __expf(Mp[idx] - M);
            L   += Lp[idx] * w;
            acc += Op[idx * D_DIM + d] * w;
        }
        out[(size_t)row * N_DIM + h * D_DIM + d] = acc / L;
    }
}

// ---------------------------------------------------------------------------
// Host launch
// ---------------------------------------------------------------------------
extern "C" void kernel_launch(void* const* d_in, const int* in_sizes, int n_in,
                              void* d_out, int out_size, void* d_ws, size_t ws_size,
                              hipStream_t stream) {
    (void)in_sizes; (void)n_in; (void)out_size; (void)ws_size;
    const float* X  = (const float*)d_in[0];
    const float* Wq = (const float*)d_in[1];
    const float* Wk = (const float*)d_in[2];
    const float* Wv = (const float*)d_in[3];
    const float* cK = (const float*)d_in[4];
    const float* cV = (const float*)d_in[5];
    float* out = (float*)d_out;

    float* ws = (float*)d_ws;
    float* Xn = ws;                 // 16*4096          = 65536
    float* Qw = ws + 65536;         // 16*4096
    float* Kw = ws + 131072;        // 16*4096
    float* Vw = ws + 196608;        // 16*4096
    float* Op = ws + 262144;        // 32*8*16*128      = 524288
    float* Mp = ws + 786432;        // 32*8*16          = 4096
    float* Lp = ws + 790528;        // 32*8*16          = 4096
                                    // total ~3.03 MB of d_ws

    rmsnorm_kernel<<<M_DIM, 256, 0, stream>>>(X, Xn);
    qkv_kernel<<<dim3(N_DIM / 64, 3), 128, 0, stream>>>(Xn, Wq, Wk, Wv, Qw, Kw, Vw);
    attn_partial_kernel<<<dim3(CHUNKS, H_DIM), 32, 0, stream>>>(Qw, Kw, Vw, cK, cV, Op, Mp, Lp);
    combine_kernel<<<H_DIM, 128, 0, stream>>>(Op, Mp, Lp, out);
}